// FaultAttention_27547920236976
// MI455X (gfx1250) — compile-verified
//
#include <hip/hip_runtime.h>
#include <math.h>

// Problem constants (match reference)
#define BB   2
#define SS   2048
#define INC  64
#define HID  512
#define NHD  8
#define HDD  64
#define RR   64
#define TKK  16
#define C1N  171
#define C2N  171
#define C3N  170
#define CHUNK 512   // key chunk for sparse scores (16 x 512 f32 = 32KB LDS)

typedef float v2f __attribute__((ext_vector_type(2)));
typedef float v8f __attribute__((ext_vector_type(8)));

__device__ __forceinline__ v8f wmma_f32_16x16x4(v2f a, v2f b, v8f c) {
  // (neg_a, A, neg_b, B, c_mod, C, reuse_a, reuse_b)
  return __builtin_amdgcn_wmma_f32_16x16x4_f32(false, a, false, b, (short)0, c,
                                               false, false);
}

// ---------------------------------------------------------------------------
// 1) Fused 3x conv1d (k=3/5/7, same-pad) + ReLU + channel concat -> h (B,S,512)
// ---------------------------------------------------------------------------
__global__ __launch_bounds__(256) void conv_concat_kernel(
    const float* __restrict__ x,
    const float* __restrict__ w1, const float* __restrict__ b1,
    const float* __restrict__ w2, const float* __restrict__ b2,
    const float* __restrict__ w3, const float* __restrict__ b3,
    float* __restrict__ h)
{
  const int row = blockIdx.x;           // b*S + s
  const int b   = row / SS;
  const int s   = row % SS;
  for (int cc = 0; cc < 2; ++cc) {
    const int c = threadIdx.x + cc * 256;
    int ksz, pad, co;
    const float* w; const float* bias;
    if (c < C1N)              { ksz = 3; pad = 1; co = c;          w = w1; bias = b1; }
    else if (c < C1N + C2N)   { ksz = 5; pad = 2; co = c - C1N;    w = w2; bias = b2; }
    else                      { ksz = 7; pad = 3; co = c - C1N-C2N;w = w3; bias = b3; }
    const float* wr = w + (size_t)co * INC * ksz;   // OIH layout: [co][ci][t]
    float acc = bias[co];
    for (int t = 0; t < ksz; ++t) {
      const int j = s + t - pad;
      if (j >= 0 && j < SS) {
        const float* xr = x + ((size_t)(b * SS + j)) * INC;
        #pragma unroll 8
        for (int ci = 0; ci < INC; ++ci)
          acc += xr[ci] * wr[ci * ksz + t];
      }
    }
    h[(size_t)row * HID + c] = acc > 0.f ? acc : 0.f;
  }
}

// ---------------------------------------------------------------------------
// 2) Generic fp32 WMMA GEMM:  O = A * W^T + bias [+ resid]
// Register-blocked 16x64 per wave (4 accumulators). K is processed in blocks
// of 16: phase 1 issues all 20 loads (4 A-frags + 16 B-frags) as a clause,
// phase 2 issues the 16 v_wmma — waits stagger across the burst, hiding
// load latency. Requires Ntiles % 4 == 0 and K % 16 == 0.
// Lane fragment mapping (wave32):
//   A: a[i] = A[mt*16 + lane%16][kk + 2*(lane/16) + i]   (contiguous float2)
//   B: b[i] = W[n   ][kk + 2*(lane/16) + i]              (contiguous float2)
//   D: c[v] = D[v + 8*(lane/16)][lane%16]
// ---------------------------------------------------------------------------
__global__ __launch_bounds__(256) void wmma_gemm_bias(
    const float* __restrict__ A, int lda, int aZOff,
    const float* __restrict__ W, int ldw,
    const float* __restrict__ bias,
    const float* __restrict__ resid,
    float* __restrict__ O, int ldo, int oZOff,
    int M, int Ntiles, int K)
{
  const int wid  = threadIdx.x >> 5;
  const int lane = threadIdx.x & 31;
  const int hl   = lane >> 4;       // lane half
  const int l16  = lane & 15;
  const int Mtiles = M >> 4;
  const int NG     = Ntiles >> 2;   // groups of 4 n-tiles (64 cols)
  const int total  = Mtiles * NG;
  const int t = blockIdx.x * (blockDim.x >> 5) + wid;
  if (t >= total) return;           // wave-uniform: EXEC stays all-ones for WMMA
  const int mt = t / NG;
  const int ng = t % NG;
  const int acol = blockIdx.z * aZOff;
  const int ocol = blockIdx.z * oZOff;

  const float* Arow = A + (size_t)(mt * 16 + l16) * lda + acol + 2 * hl;
  const float* Wr[4];
  #pragma unroll
  for (int j = 0; j < 4; ++j)
    Wr[j] = W + (size_t)(ng * 64 + j * 16 + l16) * ldw + 2 * hl;

  v8f c[4] = {};
  for (int kk = 0; kk < K; kk += 16) {
    // phase 1: all loads for this k-block
    v2f a4[4];
    v2f b4[4][4];
    #pragma unroll
    for (int u = 0; u < 4; ++u) {
      a4[u] = *(const v2f*)(Arow + kk + 4 * u);
      #pragma unroll
      for (int j = 0; j < 4; ++j)
        b4[u][j] = *(const v2f*)(Wr[j] + kk + 4 * u);
    }
    // phase 2: 16 wmma consuming the staged fragments
    #pragma unroll
    for (int u = 0; u < 4; ++u)
      #pragma unroll
      for (int j = 0; j < 4; ++j)
        c[j] = wmma_f32_16x16x4(a4[u], b4[u][j], c[j]);
  }

  #pragma unroll
  for (int j = 0; j < 4; ++j) {
    const int col = ng * 64 + j * 16 + l16;
    const float bc = bias[col];
    #pragma unroll
    for (int v = 0; v < 8; ++v) {
      const int r = mt * 16 + v + 8 * hl;
      const size_t oi = (size_t)r * ldo + ocol + col;
      float val = c[j][v] + bc;
      if (resid) val += resid[oi];
      O[oi] = val;
    }
  }
}

// ---------------------------------------------------------------------------
// 3) Local windowed attention (WIN=5). One wave per (b,h,s). Writes `local`
//    into mix buffer in (B,S,HID) layout (sparse kernel later adds sp, *0.5).
// ---------------------------------------------------------------------------
__global__ __launch_bounds__(256) void local_attn_kernel(
    const float* __restrict__ Q, const float* __restrict__ K,
    const float* __restrict__ V, float* __restrict__ mix)
{
  const int gw   = blockIdx.x * (blockDim.x >> 5) + (threadIdx.x >> 5);
  const int lane = threadIdx.x & 31;
  const int s = gw % SS;
  const int hh = (gw / SS) % NHD;
  const int b = gw / (SS * NHD);

  const float* q = Q + ((size_t)(b * SS + s)) * HID + hh * HDD;

  float sc[5];
  #pragma unroll
  for (int w = 0; w < 5; ++w) {
    const int j = s + w - 2;
    float p = 0.f;
    const bool ok = (j >= 0) && (j < SS);
    if (ok) {
      const float* kr = K + ((size_t)(b * SS + j)) * HID + hh * HDD;
      p = q[lane] * kr[lane] + q[lane + 32] * kr[lane + 32];
    }
    #pragma unroll
    for (int off = 16; off; off >>= 1) p += __shfl_xor(p, off, 32);
    sc[w] = ok ? p * 0.125f : -INFINITY;     // 1/sqrt(64)
  }
  float m = sc[0];
  #pragma unroll
  for (int w = 1; w < 5; ++w) m = fmaxf(m, sc[w]);
  float e[5], sum = 0.f;
  #pragma unroll
  for (int w = 0; w < 5; ++w) {
    e[w] = (sc[w] == -INFINITY) ? 0.f : __expf(sc[w] - m);
    sum += e[w];
  }
  const float inv = 1.f / sum;
  float o0 = 0.f, o1 = 0.f;
  #pragma unroll
  for (int w = 0; w < 5; ++w) {
    int j = s + w - 2; j = j < 0 ? 0 : (j >= SS ? SS - 1 : j);
    const float* vr = V + ((size_t)(b * SS + j)) * HID + hh * HDD;
    const float pw = e[w] * inv;
    o0 += pw * vr[lane];
    o1 += pw * vr[lane + 32];
  }
  float* o = mix + ((size_t)(b * SS + s)) * HID + hh * HDD;
  o[lane] = o0;
  o[lane + 32] = o1;
}

// ---------------------------------------------------------------------------
// 4) Sparse global path: per block = (16 queries, head hh, batch bb).
//    a) scores Q(16x64)*Kp^T via WMMA into LDS, 512 keys per chunk; each wave
//       owns 4 consecutive key tiles; k processed in blocks of 16 with
//       staged loads for latency hiding
//    b) streaming top-16 per query (insertion, stable -> matches lax.top_k)
//    c) softmax(top) -> weights; sp = sum tw * Vp[topidx]
//    d) sp2 = sp @ Wvp^T + bvp ; mix = (local + sp2) * 0.5
// ---------------------------------------------------------------------------
__global__ __launch_bounds__(256) void sparse_topk_kernel(
    const float* __restrict__ Q, const float* __restrict__ Kp,
    const float* __restrict__ Vp,
    const float* __restrict__ Wvp, const float* __restrict__ bvp,
    float* __restrict__ mix)
{
  __shared__ float sc[16 * CHUNK];
  __shared__ float topv[16][TKK];
  __shared__ int   topi[16][TKK];
  __shared__ float sp[16 * RR];

  const int qt = blockIdx.x;            // query tile
  const int hh = blockIdx.y;            // head
  const int bb = blockIdx.z;            // batch
  const int tid  = threadIdx.x;
  const int wid  = tid >> 5;
  const int lane = tid & 31;
  const int hl   = lane >> 4;
  const int l16  = lane & 15;
  const int qbase = qt * 16;

  const float* Qrow = Q + ((size_t)(bb * SS + qbase + l16)) * HID + hh * HDD + 2 * hl;

  if (tid < 16) {
    #pragma unroll
    for (int j = 0; j < TKK; ++j) { topv[tid][j] = -INFINITY; topi[tid][j] = 0; }
  }
  __syncthreads();

  for (int cb = 0; cb < SS; cb += CHUNK) {
    // --- scores for this chunk: 32 key-tiles, wave owns tiles [wid*4, wid*4+3]
    {
      const int kt0 = wid * 4;
      const float* Krow[4];
      #pragma unroll
      for (int j = 0; j < 4; ++j)
        Krow[j] = Kp + ((size_t)(bb * SS + cb + (kt0 + j) * 16 + l16)) * HID
                     + hh * HDD + 2 * hl;
      v8f c[4] = {};
      #pragma unroll
      for (int kk = 0; kk < RR; kk += 16) {
        v2f a4[4];
        v2f b4[4][4];
        #pragma unroll
        for (int u = 0; u < 4; ++u) {
          a4[u] = *(const v2f*)(Qrow + kk + 4 * u);
          #pragma unroll
          for (int j = 0; j < 4; ++j)
            b4[u][j] = *(const v2f*)(Krow[j] + kk + 4 * u);
        }
        #pragma unroll
        for (int u = 0; u < 4; ++u)
          #pragma unroll
          for (int j = 0; j < 4; ++j)
            c[j] = wmma_f32_16x16x4(a4[u], b4[u][j], c[j]);
      }
      #pragma unroll
      for (int j = 0; j < 4; ++j)
        #pragma unroll
        for (int v = 0; v < 8; ++v) {
          const int qq = v + 8 * hl;
          sc[qq * CHUNK + (kt0 + j) * 16 + l16] = c[j][v] * 0.125f; // /sqrt(R)
        }
    }
    // prefetch next chunk of Kp into cache while top-k scan runs
    if (cb + CHUNK < SS) {
      const int key = cb + CHUNK + tid;
      __builtin_prefetch(Kp + ((size_t)(bb * SS + key)) * HID + hh * HDD, 0, 0);
      __builtin_prefetch(Kp + ((size_t)(bb * SS + key + 256)) * HID + hh * HDD, 0, 0);
    }
    __syncthreads();
    // --- streaming top-16 maintenance (one thread per query) ---
    if (tid < 16) {
      const int q = tid;
      for (int k = 0; k < CHUNK; ++k) {
        const float v = sc[q * CHUNK + k];
        if (v > topv[q][TKK - 1]) {
          int p = TKK - 1;
          while (p > 0 && topv[q][p - 1] < v) {
            topv[q][p] = topv[q][p - 1];
            topi[q][p] = topi[q][p - 1];
            --p;
          }
          topv[q][p] = v;
          topi[q][p] = cb + k;
        }
      }
    }
    __syncthreads();
  }

  // --- softmax over top-16 (weights overwrite topv) ---
  if (tid < 16) {
    const int q = tid;
    const float m = topv[q][0];
    float sum = 0.f;
    #pragma unroll
    for (int j = 0; j < TKK; ++j) { const float e = __expf(topv[q][j] - m); topv[q][j] = e; sum += e; }
    const float inv = 1.f / sum;
    #pragma unroll
    for (int j = 0; j < TKK; ++j) topv[q][j] *= inv;
  }
  __syncthreads();

  // --- sp[q][r] = sum_j tw[q][j] * Vp[top_j][r] ---
  {
    const int q = tid >> 4, rg = tid & 15;
    #pragma unroll
    for (int rr = 0; rr < 4; ++rr) {
      const int r = rg + rr * 16;
      float acc = 0.f;
      #pragma unroll
      for (int j = 0; j < TKK; ++j)
        acc += topv[q][j] * Vp[((size_t)(bb * SS + topi[q][j])) * HID + hh * HDD + r];
      sp[q * RR + r] = acc;
    }
  }
  __syncthreads();

  // --- sp2 = sp @ Wvp^T + bvp ; mix = (local + sp2) * 0.5 ---
  {
    const int q = tid >> 4, jg = tid & 15;
    #pragma unroll
    for (int jj = 0; jj < 4; ++jj) {
      const int j = jg + jj * 16;
      float acc = bvp[j];
      #pragma unroll 8
      for (int r = 0; r < RR; ++r) acc += sp[q * RR + r] * Wvp[j * RR + r];
      const size_t oi = ((size_t)(bb * SS + qbase + q)) * HID + hh * HDD + j;
      mix[oi] = (mix[oi] + acc) * 0.5f;
    }
  }
}

// ---------------------------------------------------------------------------
// 5) LayerNorm over HID=512 per row. One block per row, 256 threads.
// ---------------------------------------------------------------------------
__global__ __launch_bounds__(256) void ln_kernel(
    const float* __restrict__ res,
    const float* __restrict__ g, const float* __restrict__ bl,
    float* __restrict__ out)
{
  __shared__ float sh1[256], sh2[256];
  const int row = blockIdx.x;
  const int tid = threadIdx.x;
  const float* r = res + (size_t)row * HID;
  const float v0 = r[tid], v1 = r[tid + 256];
  sh1[tid] = v0 + v1;
  sh2[tid] = v0 * v0 + v1 * v1;
  __syncthreads();
  for (int off = 128; off; off >>= 1) {
    if (tid < off) { sh1[tid] += sh1[tid + off]; sh2[tid] += sh2[tid + off]; }
    __syncthreads();
  }
  const float mu  = sh1[0] * (1.f / HID);
  const float var = sh2[0] * (1.f / HID) - mu * mu;
  const float inv = rsqrtf(var + 1e-5f);
  float* o = out + (size_t)row * HID;
  o[tid]       = (v0 - mu) * inv * g[tid]       + bl[tid];
  o[tid + 256] = (v1 - mu) * inv * g[tid + 256] + bl[tid + 256];
}

// ---------------------------------------------------------------------------
extern "C" void kernel_launch(void* const* d_in, const int* in_sizes, int n_in,
                              void* d_out, int out_size, void* d_ws, size_t ws_size,
                              hipStream_t stream) {
  const float* x   = (const float*)d_in[0];
  const float* w1  = (const float*)d_in[1];
  const float* b1  = (const float*)d_in[2];
  const float* w2  = (const float*)d_in[3];
  const float* b2  = (const float*)d_in[4];
  const float* w3  = (const float*)d_in[5];
  const float* b3  = (const float*)d_in[6];
  const float* Wq  = (const float*)d_in[7];
  const float* bq  = (const float*)d_in[8];
  const float* Wk  = (const float*)d_in[9];
  const float* bk  = (const float*)d_in[10];
  const float* Wv  = (const float*)d_in[11];
  const float* bv  = (const float*)d_in[12];
  const float* Wkp = (const float*)d_in[13];
  const float* bkp = (const float*)d_in[14];
  const float* Wvp = (const float*)d_in[15];
  const float* bvp = (const float*)d_in[16];
  const float* Wo  = (const float*)d_in[17];
  const float* bo  = (const float*)d_in[18];
  const float* lng = (const float*)d_in[19];
  const float* lnb = (const float*)d_in[20];

  const size_t NE = (size_t)BB * SS * HID;   // 2*2048*512 = 2,097,152 floats
  float* h    = (float*)d_ws;
  float* Qb   = h    + NE;
  float* Kb   = Qb   + NE;
  float* Vb   = Kb   + NE;
  float* Kpb  = Vb   + NE;
  float* Vpb  = Kpb  + NE;
  float* mixb = Vpb  + NE;
  float* resb = mixb + NE;

  const int M = BB * SS;                     // 4096 rows

  // 1) convs -> h
  conv_concat_kernel<<<M, 256, 0, stream>>>(x, w1, b1, w2, b2, w3, b3, h);

  // 2) QKV projections (M=4096, N=512, K=512):
  //    256 Mtiles * 8 Ngroups = 2048 waves -> 256 blocks of 8 waves
  wmma_gemm_bias<<<256, 256, 0, stream>>>(h, HID, 0, Wq, HID, bq, nullptr,
                                          Qb, HID, 0, M, HID / 16, HID);
  wmma_gemm_bias<<<256, 256, 0, stream>>>(h, HID, 0, Wk, HID, bk, nullptr,
                                          Kb, HID, 0, M, HID / 16, HID);
  wmma_gemm_bias<<<256, 256, 0, stream>>>(h, HID, 0, Wv, HID, bv, nullptr,
                                          Vb, HID, 0, M, HID / 16, HID);

  // 3) per-head Kp/Vp (M=4096, N=64, K=64), grid.z = heads
  //    256 Mtiles * 1 Ngroup = 256 waves -> 32 blocks per head
  {
    dim3 g(32, 1, NHD);
    wmma_gemm_bias<<<g, 256, 0, stream>>>(Kb, HID, HDD, Wkp, HDD, bkp, nullptr,
                                          Kpb, HID, RR, M, RR / 16, HDD);
    wmma_gemm_bias<<<g, 256, 0, stream>>>(Vb, HID, HDD, Wvp, HDD, bvp, nullptr,
                                          Vpb, HID, RR, M, RR / 16, HDD);
  }

  // 4) local windowed attention -> mix (writes full buffer)
  local_attn_kernel<<<(BB * NHD * SS) / 8, 256, 0, stream>>>(Qb, Kb, Vb, mixb);

  // 5) sparse top-k path; mix = (local + sp)*0.5
  {
    dim3 g(SS / 16, NHD, BB);
    sparse_topk_kernel<<<g, 256, 0, stream>>>(Qb, Kpb, Vpb, Wvp, bvp, mixb);
  }

  // 6) output projection + residual h
  wmma_gemm_bias<<<256, 256, 0, stream>>>(mixb, HID, 0, Wo, HID, bo, h,
                                          resb, HID, 0, M, HID / 16, HID);

  // 7) LayerNorm -> d_out
  ln_kernel<<<M, 256, 0, stream>>>(resb, lng, lnb, (float*)d_out);
}